// PriorityWeightingLossPAM1_62629213110515
// MI455X (gfx1250) — compile-verified
//
#include <hip/hip_runtime.h>
#include <hip/hip_bf16.h>
#include <math.h>

// ---------------- problem constants ----------------
#define NFFT        512
#define HOP         480
#define NBINS       257
#define NPADF       272           // 257 padded to 17*16
#define NTILES      (NPADF / 16)  // 17
#define TFRAMES     184           // 1 + 88200/480
#define BATCH       8
#define LSIG        88200
#define FRAMES_PER_SIG (BATCH * TFRAMES)   // 1472
#define TOTAL_FRAMES   (2 * FRAMES_PER_SIG) // 2944
#define MTILES         (TOTAL_FRAMES / 16)  // 184
#define NCOLS          (BATCH * TFRAMES)    // 1472 (b,t) columns
#define LDS_STRIDE  516           // 512 + 4 pad -> conflict-free WMMA A reads
#define PAIR_ROW    (2 * NPADF)   // 544 floats per k-pair row in twiddle tables
#define TWO_PI      6.28318530717958647692f

typedef float v2f __attribute__((ext_vector_type(2)));
typedef float v8f __attribute__((ext_vector_type(8)));

// ---------------- kernel 0: DFT twiddle tables (k-pair interleaved) + window ----
// Layout: w2[(k>>1)*544 + 2*f + (k&1)] -> per-lane B fragment is one b64 load.
__global__ __launch_bounds__(256) void pam_tables_dft(float* __restrict__ wcos2,
                                                      float* __restrict__ wsin2,
                                                      float* __restrict__ window) {
  int idx = blockIdx.x * blockDim.x + threadIdx.x;
  if (idx < NFFT) {
    // periodic hann: 0.5*(1 - cos(2*pi*n/N))
    window[idx] = 0.5f * (1.0f - cosf(TWO_PI * (float)idx / (float)NFFT));
  }
  if (idx < NFFT * NPADF) {
    int k = idx / NPADF;
    int f = idx - k * NPADF;
    float c = 0.0f, s = 0.0f;
    if (f < NBINS) {
      // exact integer phase reduction: angle = 2*pi*((k*f) mod 512)/512
      int ph = (k * f) & (NFFT - 1);
      float ang = TWO_PI * (float)ph / (float)NFFT;
      c = cosf(ang);
      s = sinf(ang);   // sign irrelevant: power = re^2 + im^2
    }
    int o = (k >> 1) * PAIR_ROW + 2 * f + (k & 1);
    wcos2[o] = c;
    wsin2[o] = s;
  }
}

// ---------------- kernel 1: ATH + transposed spreading matrix ----------------
__device__ __forceinline__ float pam_bark(int f) {
  float freq = (float)f * (44100.0f / (float)NFFT);
  float r = freq / 7500.0f;
  return 13.0f * atanf(0.00076f * freq) + 3.5f * atanf(r * r);
}

__global__ __launch_bounds__(256) void pam_tables_psy(float* __restrict__ ath,
                                                      float* __restrict__ spreadT) {
  int idx = blockIdx.x * blockDim.x + threadIdx.x;
  if (idx < NBINS) {
    float freq = (float)idx * (44100.0f / (float)NFFT);
    float fk = fmaxf(freq, 1e-6f) * 1e-3f;   // kHz
    float d = fk - 3.3f;
    float ath_db = 3.64f * powf(fk, -0.8f)
                 - 6.5f * expf(-0.6f * d * d)
                 + 0.001f * fk * fk * fk * fk;
    ath_db = fminf(fmaxf(ath_db, -100.0f), 100.0f);
    ath[idx] = fmaxf(exp10f(ath_db * 0.1f), 1e-12f);
  }
  if (idx < NBINS * NBINS) {
    // spreadT[fm][f] == SPREAD[f][fm], contiguous in f for coalesced column reads
    int fm = idx / NBINS;
    int f  = idx - fm * NBINS;
    float df = pam_bark(f) - pam_bark(fm);
    float sdb = (df >= 0.0f) ? (-17.0f * df) : (27.0f * df); // -27*|df| for df<0
    spreadT[idx] = exp10f(sdb * 0.1f);
  }
}

// ---------------- kernel 2: STFT power via WMMA f32 16x16x4 GEMM ----------------
// M = 2944 frames (pred then target), K = 512 samples, N = 272 padded bins.
// One block per 16-frame M-tile; 4 waves sweep the 17 N-tiles; frames staged in LDS.
__global__ __launch_bounds__(128) void pam_stft_wmma(const float* __restrict__ pred,
                                                     const float* __restrict__ target,
                                                     const float* __restrict__ wcos2,
                                                     const float* __restrict__ wsin2,
                                                     const float* __restrict__ window,
                                                     float* __restrict__ powerbuf) {
  __shared__ float fr[16 * LDS_STRIDE];

  const int mtile = blockIdx.x;
  const int tid   = threadIdx.x;

  // Block-uniform frame decomposition (tile never crosses the signal boundary:
  // 1472 % 16 == 0). SALU divides once per block instead of per element.
  const int gm0  = mtile * 16;
  const int sig  = gm0 / FRAMES_PER_SIG;
  const int rem0 = gm0 - sig * FRAMES_PER_SIG;
  const int b0   = rem0 / TFRAMES;
  const int t0   = rem0 - b0 * TFRAMES;   // t0 + 15 may wrap b at most once

  // Cooperative load of 16 windowed frames (reflect-padded) into LDS.
  const float* __restrict__ src = sig ? target : pred;
  for (int i = tid; i < 16 * NFFT; i += 128) {
    int m = i >> 9;
    int n = i & (NFFT - 1);
    int t = t0 + m;
    int b = b0;
    if (t >= TFRAMES) { t -= TFRAMES; b += 1; }
    int s = t * HOP + n - (NFFT / 2);
    s = (s < 0) ? -s : s;
    s = (s >= LSIG) ? (2 * LSIG - 2 - s) : s;
    fr[m * LDS_STRIDE + n] = src[b * LSIG + s] * window[n];
  }
  __syncthreads();

  const int wave = tid >> 5;
  const int lane = tid & 31;
  const int mrow = lane & 15;
  const int h    = lane >> 4;            // lane-half selects K sub-pair
  const float* __restrict__ arow = &fr[mrow * LDS_STRIDE];

  for (int ntile = wave; ntile < NTILES; ntile += 4) {
    const int ncol = ntile * 16 + (lane & 15);
    // per-lane base into the pair-interleaved twiddle tables
    const float* __restrict__ pcos = wcos2 + 2 * ncol + h * PAIR_ROW;
    const float* __restrict__ psin = wsin2 + 2 * ncol + h * PAIR_ROW;
    v8f ccos = {};
    v8f csin = {};
    for (int k0 = 0; k0 < NFFT; k0 += 4) {
      const int kb = k0 + 2 * h;
      v2f a  = *(const v2f*)(arow + kb);                       // ds b64
      v2f bc = *(const v2f*)(pcos + (size_t)(k0 >> 1) * PAIR_ROW); // global b64
      v2f bs = *(const v2f*)(psin + (size_t)(k0 >> 1) * PAIR_ROW); // global b64
      // D = A(16x4,f32) * B(4x16,f32) + C : full fp32 precision DFT accumulate
      ccos = __builtin_amdgcn_wmma_f32_16x16x4_f32(false, a, false, bc, (short)0, ccos, false, false);
      csin = __builtin_amdgcn_wmma_f32_16x16x4_f32(false, a, false, bs, (short)0, csin, false, false);
    }
    const int f = ntile * 16 + (lane & 15);
    if (f < NBINS) {
#pragma unroll
      for (int r = 0; r < 8; ++r) {
        int m = r + 8 * h;                 // C/D layout: VGPR r -> M = r + 8*lanehalf
        int t = t0 + m;
        int b = b0;
        if (t >= TFRAMES) { t -= TFRAMES; b += 1; }
        float pw = ccos[r] * ccos[r] + csin[r] * csin[r] + 1e-12f;
        // [sig][b][t][f] layout: lanes 0..15 write contiguous f
        powerbuf[(((size_t)sig * BATCH + b) * TFRAMES + t) * NBINS + f] = pw;
      }
    }
  }
}

// ---------------- kernel 3: per-(b,t) masking + weighted error ----------------
__global__ __launch_bounds__(256) void pam_loss(const float* __restrict__ powerbuf,
                                                const float* __restrict__ ath,
                                                const float* __restrict__ spreadT,
                                                float* __restrict__ partial) {
  const int bt = blockIdx.x;               // 0 .. NCOLS-1 == (b*TFRAMES + t)

  __shared__ float pt[NBINS];
  __shared__ float pp[NBINS];
  __shared__ int   midx[NBINS];
  __shared__ float mpow[NBINS];
  __shared__ float red[256];
  __shared__ int   mcount;

  const float* __restrict__ prow = powerbuf + (size_t)bt * NBINS;                      // pred row
  const float* __restrict__ trow = powerbuf + ((size_t)NCOLS + bt) * NBINS;            // target row

  if (threadIdx.x == 0) mcount = 0;
  for (int f = threadIdx.x; f < NBINS; f += 256) {   // fully coalesced rows
    pp[f] = prow[f];
    pt[f] = trow[f];
  }
  __syncthreads();

  // tonal peaks: strict local max in power domain (== dB domain), level >= -40dB <=> p >= 1e-4
  for (int f = threadIdx.x; f < NBINS; f += 256) {
    if (f >= 1 && f <= NBINS - 2) {
      float c = pt[f];
      if (c > pt[f - 1] && c > pt[f + 1] && c >= 1e-4f) {
        int j = atomicAdd(&mcount, 1);   // order-independent: feeds a max
        midx[j] = f;
        mpow[j] = c;
      }
    }
  }
  __syncthreads();

  const int cnt = mcount;
  float acc = 0.0f;
  for (int f = threadIdx.x; f < NBINS; f += 256) {
    float masking = 0.0f;
    for (int j = 0; j < cnt; ++j)
      masking = fmaxf(masking, mpow[j] * spreadT[midx[j] * NBINS + f]);
    float combined = masking + ath[f];
    float w  = log10f(pt[f] / (combined + 1e-12f) + 1.0f);
    float mp = sqrtf(pp[f] + 1e-12f);
    float mt = sqrtf(pt[f] + 1e-12f);
    float d  = mp - mt;
    acc += w * d * d;
  }

  red[threadIdx.x] = acc;
  __syncthreads();
  for (int s = 128; s > 0; s >>= 1) {
    if (threadIdx.x < s) red[threadIdx.x] += red[threadIdx.x + s];
    __syncthreads();
  }
  if (threadIdx.x == 0) partial[bt] = red[0];
}

// ---------------- kernel 4: deterministic final reduction ----------------
__global__ __launch_bounds__(256) void pam_reduce(const float* __restrict__ partial,
                                                  float* __restrict__ out) {
  __shared__ float red[256];
  float acc = 0.0f;
  for (int i = threadIdx.x; i < NCOLS; i += 256) acc += partial[i];
  red[threadIdx.x] = acc;
  __syncthreads();
  for (int s = 128; s > 0; s >>= 1) {
    if (threadIdx.x < s) red[threadIdx.x] += red[threadIdx.x + s];
    __syncthreads();
  }
  if (threadIdx.x == 0)
    out[0] = red[0] * (1.0f / (float)(BATCH * NBINS * TFRAMES));
}

// ---------------- host launch ----------------
extern "C" void kernel_launch(void* const* d_in, const int* in_sizes, int n_in,
                              void* d_out, int out_size, void* d_ws, size_t ws_size,
                              hipStream_t stream) {
  (void)in_sizes; (void)n_in; (void)out_size; (void)ws_size;
  const float* pred   = (const float*)d_in[0];
  const float* target = (const float*)d_in[1];
  float* out = (float*)d_out;

  // workspace layout (floats)
  float* ws       = (float*)d_ws;
  float* wcos2    = ws;                                   // 512*272 = 139264
  float* wsin2    = wcos2 + NFFT * NPADF;                 // 139264
  float* window   = wsin2 + NFFT * NPADF;                 // 512
  float* ath      = window + NFFT;                        // 257 (padded 320)
  float* spreadT  = ath + 320;                            // 66049 (padded 66304)
  float* powerbuf = spreadT + 66304;                      // 2*8*184*257 = 756608
  float* partial  = powerbuf + (size_t)2 * BATCH * TFRAMES * NBINS; // 1472
  // total ~4.42 MB of workspace

  pam_tables_dft<<<(NFFT * NPADF + 255) / 256, 256, 0, stream>>>(wcos2, wsin2, window);
  pam_tables_psy<<<(NBINS * NBINS + 255) / 256, 256, 0, stream>>>(ath, spreadT);
  pam_stft_wmma<<<MTILES, 128, 0, stream>>>(pred, target, wcos2, wsin2, window, powerbuf);
  pam_loss<<<NCOLS, 256, 0, stream>>>(powerbuf, ath, spreadT, partial);
  pam_reduce<<<1, 256, 0, stream>>>(partial, out);
}